// TargetNetwork_20985210208628
// MI455X (gfx1250) — compile-verified
//
#include <hip/hip_runtime.h>

typedef __attribute__((ext_vector_type(2))) float v2f;
typedef __attribute__((ext_vector_type(8))) float v8f;

#define BB 4096
#define DD 128
#define HH 256
#define CC 64

// Layer 1: h[b, h0+m] = relu( sum_d x[b,d] * w1[b,d,h0+m] + b1[b,h0+m] )
// One wave per (b, h-tile of 16). A = w1^T tile (16x4), B = x chunk replicated over N.
__global__ __launch_bounds__(256) void l1_wmma(
    const float* __restrict__ x, const float* __restrict__ w1,
    const float* __restrict__ b1, float* __restrict__ hbuf)
{
    const int lane        = threadIdx.x & 31;
    const int wave_in_blk = threadIdx.x >> 5;
    const int wave        = blockIdx.x * 8 + wave_in_blk;
    const int b           = wave >> 4;          // 16 h-tiles per sample
    const int h0          = (wave & 15) << 4;

    const int half = lane >> 4;                 // 0 -> K={0,1}, 1 -> K={2,3}
    const int hcol = lane & 15;                 // M (layer-1 output feature) index

    const float* w1b = w1 + (size_t)b * (DD * HH) + h0 + hcol;
    const float* xb  = x  + (size_t)b * DD;

    v8f c = {};
#pragma unroll 4
    for (int k0 = 0; k0 < DD; k0 += 4) {
        const int d0 = k0 + (half << 1);
        v2f a, bv;
        // A (16x4 f32): vgpr0 = K=(0|2), vgpr1 = K=(1|3); streamed once -> NT hint
        a.x = __builtin_nontemporal_load(&w1b[(size_t)d0 * HH]);
        a.y = __builtin_nontemporal_load(&w1b[(size_t)(d0 + 1) * HH]);
        // B (4x16 f32): replicate x[k] across all N lanes of each half
        bv = *(const v2f*)(xb + d0);            // 8B-aligned (d0 even)
        c = __builtin_amdgcn_wmma_f32_16x16x4_f32(false, a, false, bv,
                                                  (short)0, c, false, false);
    }

    // All N columns hold identical GEMV results; lane 0 has M=0..7, lane 16 has M=8..15.
    if (hcol == 0) {
        const int mbase = half << 3;
        float*       ho = hbuf + (size_t)b * HH + h0 + mbase;
        const float* bb = b1   + (size_t)b * HH + h0 + mbase;
#pragma unroll
        for (int r = 0; r < 8; ++r) {
            float v = c[r] + bb[r];
            ho[r] = v > 0.0f ? v : 0.0f;
        }
    }
}

// Layer 2: out[b, c0+m] = sum_h h[b,h] * w2[b,h,c0+m] + b2[b,c0+m]
__global__ __launch_bounds__(256) void l2_wmma(
    const float* __restrict__ hbuf, const float* __restrict__ w2,
    const float* __restrict__ b2, float* __restrict__ out)
{
    const int lane        = threadIdx.x & 31;
    const int wave_in_blk = threadIdx.x >> 5;
    const int wave        = blockIdx.x * 8 + wave_in_blk;
    const int b           = wave >> 2;          // 4 c-tiles per sample
    const int c0          = (wave & 3) << 4;

    const int half = lane >> 4;
    const int ccol = lane & 15;

    const float* w2b = w2   + (size_t)b * (HH * CC) + c0 + ccol;
    const float* hb  = hbuf + (size_t)b * HH;

    v8f c = {};
#pragma unroll 4
    for (int k0 = 0; k0 < HH; k0 += 4) {
        const int d0 = k0 + (half << 1);
        v2f a, bv;
        a.x = __builtin_nontemporal_load(&w2b[(size_t)d0 * CC]);
        a.y = __builtin_nontemporal_load(&w2b[(size_t)(d0 + 1) * CC]);
        bv = *(const v2f*)(hb + d0);
        c = __builtin_amdgcn_wmma_f32_16x16x4_f32(false, a, false, bv,
                                                  (short)0, c, false, false);
    }

    if (ccol == 0) {
        const int mbase = half << 3;
        float*       oo = out + (size_t)b * CC + c0 + mbase;
        const float* bb = b2  + (size_t)b * CC + c0 + mbase;
#pragma unroll
        for (int r = 0; r < 8; ++r)
            oo[r] = c[r] + bb[r];
    }
}

extern "C" void kernel_launch(void* const* d_in, const int* in_sizes, int n_in,
                              void* d_out, int out_size, void* d_ws, size_t ws_size,
                              hipStream_t stream) {
    const float* x  = (const float*)d_in[0];
    const float* w1 = (const float*)d_in[1];
    const float* b1 = (const float*)d_in[2];
    const float* w2 = (const float*)d_in[3];
    const float* b2 = (const float*)d_in[4];
    float* out  = (float*)d_out;
    float* hbuf = (float*)d_ws;   // B*H floats = 4 MB intermediate activations

    // K1: 4096 samples * 16 h-tiles = 65536 waves, 8 waves (256 thr) per block
    l1_wmma<<<(BB * (HH / 16)) / 8, 256, 0, stream>>>(x, w1, b1, hbuf);
    // K2: 4096 samples * 4 c-tiles = 16384 waves
    l2_wmma<<<(BB * (CC / 16)) / 8, 256, 0, stream>>>(hbuf, w2, b2, out);
}